// new_loss_fn_47811575939325
// MI455X (gfx1250) — compile-verified
//
#include <hip/hip_runtime.h>
#include <hip/hip_bf16.h>
#include <math.h>

// Problem constants (fixed by the reference).
#define N_ROWS 1048576
#define N_CLS  128
#define BLOCKS 1024
#define TPB    256
#define WPB    (TPB / 32)                        // 8 wave32 per block
#define ROWS_PER_WAVE (N_ROWS / (BLOCKS * WPB))  // 128 contiguous rows per wave

// Workspace layout (floats). ~1.03 MB total, every cell rewritten each call.
#define WS_PICK 0
#define WS_COL  (BLOCKS)
#define WS_CNT  (BLOCKS + BLOCKS * N_CLS)

typedef float v2f __attribute__((ext_vector_type(2)));
typedef float v8f __attribute__((ext_vector_type(8)));

// ---------------------------------------------------------------------------
// Kernel 1: HBM-bound stream. Each wave owns 128 contiguous rows; each lane
// owns 4 classes (float4 load = one full 512B row per wave). Accumulate
// exp-colsum, target counts, and picked log-prob in registers; reduce the 8
// wave partials per block with V_WMMA_F32_16X16X4_F32 (A = ones -> D[m,n] =
// sum_k B[k,n] + C[m,n], i.e. a K=4 column-sum engine), then write per-block
// partials to ws (no atomics -> deterministic).
// ---------------------------------------------------------------------------
__global__ __launch_bounds__(TPB) void loss_main(const float* __restrict__ pred,
                                                 const int* __restrict__ tgt,
                                                 float* __restrict__ ws) {
  __shared__ float sCol[WPB][N_CLS];
  __shared__ float sCnt[WPB][N_CLS];
  __shared__ float sPick[TPB];

  const int tid     = threadIdx.x;
  const int lane    = tid & 31;
  const int wave    = tid >> 5;
  const int gwave   = blockIdx.x * WPB + wave;
  const int row0    = gwave * ROWS_PER_WAVE;
  const int colBase = lane * 4;

  float cs0 = 0.f, cs1 = 0.f, cs2 = 0.f, cs3 = 0.f;   // exp column sums
  float ct0 = 0.f, ct1 = 0.f, ct2 = 0.f, ct3 = 0.f;   // target counts
  float picked = 0.f;                                  // sum pred[r, t_r]

  for (int r = row0; r < row0 + ROWS_PER_WAVE; r += 4) {
    // Stream hint: pull 8 rows ahead into cache (gfx1250 global_prefetch_b8).
    __builtin_prefetch(pred + (size_t)(r + 8) * N_CLS + colBase, 0, 1);
#pragma unroll
    for (int u = 0; u < 4; ++u) {
      const float4 v =
          *reinterpret_cast<const float4*>(pred + (size_t)(r + u) * N_CLS + colBase);
      cs0 += __expf(v.x);
      cs1 += __expf(v.y);
      cs2 += __expf(v.z);
      cs3 += __expf(v.w);
      const int j = tgt[r + u] - colBase;   // exactly one lane hits [0,4)
      if (j == 0)      { picked += v.x; ct0 += 1.f; }
      else if (j == 1) { picked += v.y; ct1 += 1.f; }
      else if (j == 2) { picked += v.z; ct2 += 1.f; }
      else if (j == 3) { picked += v.w; ct3 += 1.f; }
    }
  }

  sCol[wave][colBase + 0] = cs0; sCol[wave][colBase + 1] = cs1;
  sCol[wave][colBase + 2] = cs2; sCol[wave][colBase + 3] = cs3;
  sCnt[wave][colBase + 0] = ct0; sCnt[wave][colBase + 1] = ct1;
  sCnt[wave][colBase + 2] = ct2; sCnt[wave][colBase + 3] = ct3;
  sPick[tid] = picked;
  __syncthreads();

  // Deterministic LDS tree for the picked-logprob scalar.
  for (int s = TPB / 2; s > 0; s >>= 1) {
    if (tid < s) sPick[tid] += sPick[tid + s];
    __syncthreads();
  }
  if (tid == 0) ws[WS_PICK + blockIdx.x] = sPick[0];

  // Wave 0 (EXEC all-ones, as WMMA requires) reduces the 8x128 wave partials.
  if (tid < 32) {
    const int half = (lane >> 4) & 1;   // which 16-lane half of the wave
    const int n    = lane & 15;         // column within 16-wide tile
    v2f ones; ones.x = 1.f; ones.y = 1.f;
#pragma unroll
    for (int tile = 0; tile < 8; ++tile) {
      const int cb = tile * 16 + n;
      v8f accS = {};
      v8f accC = {};
      // Two K=4 WMMAs sum all 8 wave-partial rows per column. The 4 B slots
      // {VGPR.x/.y x lane-half} each carry one distinct row; since D[m,n] =
      // sum_k B[k,n], any slot->row bijection yields the exact column sum.
#pragma unroll
      for (int g = 0; g < 2; ++g) {
        v2f b; b.x = sCol[g * 4 + half][cb]; b.y = sCol[g * 4 + 2 + half][cb];
        accS = __builtin_amdgcn_wmma_f32_16x16x4_f32(false, ones, false, b,
                                                     (short)0, accS, false, false);
        v2f c; c.x = sCnt[g * 4 + half][cb]; c.y = sCnt[g * 4 + 2 + half][cb];
        accC = __builtin_amdgcn_wmma_f32_16x16x4_f32(false, ones, false, c,
                                                     (short)0, accC, false, false);
      }
      // D layout: VGPR0, lanes 0-15 hold row M=0, N=lane -> per-column totals.
      if (lane < 16) {
        ws[WS_COL + blockIdx.x * N_CLS + tile * 16 + lane] = accS[0];
        ws[WS_CNT + blockIdx.x * N_CLS + tile * 16 + lane] = accC[0];
      }
    }
  }
}

// ---------------------------------------------------------------------------
// Kernel 2: single block folds the 1024 per-block partials in a fixed order
// (deterministic), forms sum_c |counts_c - colsum_c|, and writes the scalar
// (diffsum - picked_sum) / N  ==  nll + extra.
// ---------------------------------------------------------------------------
__global__ __launch_bounds__(256) void loss_reduce(const float* __restrict__ ws,
                                                   float* __restrict__ out) {
  __shared__ float sDiff[N_CLS];
  __shared__ float sP;
  const int t = threadIdx.x;

  if (t < N_CLS) {
    float cs = 0.f, ct = 0.f;
    for (int b = 0; b < BLOCKS; ++b) {
      cs += ws[WS_COL + b * N_CLS + t];
      ct += ws[WS_CNT + b * N_CLS + t];
    }
    sDiff[t] = fabsf(ct - cs);
  }
  if (t == 128) {
    float p = 0.f;
    for (int b = 0; b < BLOCKS; ++b) p += ws[WS_PICK + b];
    sP = p;
  }
  __syncthreads();

  for (int s = N_CLS / 2; s > 0; s >>= 1) {
    if (t < s) sDiff[t] += sDiff[t + s];
    __syncthreads();
  }
  if (t == 0) out[0] = (sDiff[0] - sP) * (1.0f / (float)N_ROWS);
}

extern "C" void kernel_launch(void* const* d_in, const int* in_sizes, int n_in,
                              void* d_out, int out_size, void* d_ws, size_t ws_size,
                              hipStream_t stream) {
  (void)in_sizes; (void)n_in; (void)out_size; (void)ws_size;
  const float* pred = (const float*)d_in[0];
  const int*   tgt  = (const int*)d_in[1];
  float*       ws   = (float*)d_ws;
  float*       out  = (float*)d_out;

  loss_main<<<BLOCKS, TPB, 0, stream>>>(pred, tgt, ws);
  loss_reduce<<<1, 256, 0, stream>>>(ws, out);
}